// BinsChamferLoss_44994077392948
// MI455X (gfx1250) — compile-verified
//
#include <hip/hip_runtime.h>
#include <hip/hip_bf16.h>

#define MIN_DEPTH 1e-3f
#define BIGF      1e10f
#define P         256      // bins per batch == blockDim.x
#define PIX_PER_BLOCK 1024 // 4 pixels per thread

#if defined(__gfx1250__)
#define AS1 __attribute__((address_space(1)))
#define AS3 __attribute__((address_space(3)))
#endif

// ---------------------------------------------------------------- init
__global__ void chamfer_init(unsigned int* __restrict__ binmin,
                             float* __restrict__ sum_y,
                             unsigned int* __restrict__ cnt,
                             float* __restrict__ out, int n) {
    const int tid = threadIdx.x;
    const unsigned int bigbits = __float_as_uint(BIGF);
    for (int i = tid; i < n * P; i += blockDim.x) binmin[i] = bigbits;
    if (tid < n) { sum_y[tid] = 0.0f; cnt[tid] = 0u; }
    if (tid == 0) out[0] = 0.0f;
}

// ---------------------------------------------------------------- main
__global__ void __launch_bounds__(P) chamfer_main(
        const float* __restrict__ bins,     // [n, P]
        const float* __restrict__ tp,       // [n, m]
        int m,
        unsigned int* __restrict__ binmin,  // [n, P] float-bits, pre-init BIG
        float* __restrict__ sum_y,          // [n]
        unsigned int* __restrict__ cnt)     // [n]
{
    __shared__ float4       sbins4[P / 4];
    __shared__ float4       spx4[PIX_PER_BLOCK / 4];     // invalid => QNaN
    __shared__ float        ssum;
    __shared__ unsigned int scnt;

    float* sbins = (float*)sbins4;

    const int tid   = threadIdx.x;
    const int batch = blockIdx.y;
    const int base  = blockIdx.x * PIX_PER_BLOCK;
    const float QNAN = __uint_as_float(0x7fc00000u);

    if (tid == 0) { ssum = 0.0f; scnt = 0u; }

    // ---- bins -> LDS via async global->LDS (ASYNCcnt path) when available
#if defined(__gfx1250__) && __has_builtin(__builtin_amdgcn_global_load_async_to_lds_b32)
    {
        float* gsrc = const_cast<float*>(bins + batch * P + tid);
        __builtin_amdgcn_global_load_async_to_lds_b32(
            (AS1 int*)gsrc, (AS3 int*)&sbins[tid], /*offset=*/0, /*cpol=*/0);
    }
  #if __has_builtin(__builtin_amdgcn_s_wait_asynccnt)
    __builtin_amdgcn_s_wait_asynccnt(0);
  #endif
#else
    sbins[tid] = bins[batch * P + tid];
#endif

    // ---- 4 contiguous pixels per thread: one b128 load, one b128 LDS store
    float px[4];
    {
        const int idx0 = base + 4 * tid;
        const float* src = tp + (size_t)batch * m + idx0;
        if (idx0 + 3 < m) {
            const float4 v4 = *(const float4*)src;        // global_load_b128
            px[0] = v4.x; px[1] = v4.y; px[2] = v4.z; px[3] = v4.w;
        } else {
#pragma unroll
            for (int k = 0; k < 4; ++k)
                px[k] = (idx0 + k < m) ? src[k] : 0.0f;   // <MIN_DEPTH => invalid
        }
        // masked copy for Pass B: invalid pixels become QNaN (ignored by min_num)
        float4 msk;
        msk.x = (px[0] >= MIN_DEPTH) ? px[0] : QNAN;
        msk.y = (px[1] >= MIN_DEPTH) ? px[1] : QNAN;
        msk.z = (px[2] >= MIN_DEPTH) ? px[2] : QNAN;
        msk.w = (px[3] >= MIN_DEPTH) ? px[3] : QNAN;
        spx4[tid] = msk;                                  // ds_store_b128
    }
    __syncthreads();

    // ---- Pass A: per-pixel nearest bin (cham_y numerator) -----------------
    float pm0 = BIGF, pm1 = BIGF, pm2 = BIGF, pm3 = BIGF;
#pragma unroll 4
    for (int i = 0; i < P / 4; ++i) {
        const float4 b4 = sbins4[i];                      // ds_load_b128 broadcast
#pragma unroll
        for (int u = 0; u < 4; ++u) {
            const float b = (u == 0) ? b4.x : (u == 1) ? b4.y : (u == 2) ? b4.z : b4.w;
            float d0 = b - px[0]; pm0 = fminf(pm0, d0 * d0);
            float d1 = b - px[1]; pm1 = fminf(pm1, d1 * d1);
            float d2 = b - px[2]; pm2 = fminf(pm2, d2 * d2);
            float d3 = b - px[3]; pm3 = fminf(pm3, d3 * d3);
        }
    }
    float        localSum = 0.0f;
    unsigned int localCnt = 0u;
    if (px[0] >= MIN_DEPTH) { localSum += pm0; ++localCnt; }
    if (px[1] >= MIN_DEPTH) { localSum += pm1; ++localCnt; }
    if (px[2] >= MIN_DEPTH) { localSum += pm2; ++localCnt; }
    if (px[3] >= MIN_DEPTH) { localSum += pm3; ++localCnt; }

    // wave32 reduction, then LDS atomics, then one global atomic per block
#pragma unroll
    for (int off = 16; off > 0; off >>= 1) {
        localSum += __shfl_xor(localSum, off, 32);
        localCnt += __shfl_xor(localCnt, off, 32);
    }
    if ((tid & 31) == 0) {
        atomicAdd(&ssum, localSum);     // ds_add_f32
        atomicAdd(&scnt, localCnt);     // ds_add_u32
    }

    // ---- Pass B: per-bin nearest valid pixel (cham_x) ---------------------
    // invalid pixels are QNaN in LDS: (b - NaN)^2 = NaN, min_num ignores NaN.
    const float b  = sbins[tid];
    float       bm = BIGF;
#pragma unroll 4
    for (int j = 0; j < PIX_PER_BLOCK / 4; ++j) {
        const float4 v4 = spx4[j];                        // ds_load_b128 broadcast
        float d0 = b - v4.x; bm = fminf(bm, d0 * d0);
        float d1 = b - v4.y; bm = fminf(bm, d1 * d1);
        float d2 = b - v4.z; bm = fminf(bm, d2 * d2);
        float d3 = b - v4.w; bm = fminf(bm, d3 * d3);
    }
    // non-negative floats: bit pattern is order-preserving -> uint atomic min
    atomicMin(&binmin[batch * P + tid], __float_as_uint(bm));

    __syncthreads();
    if (tid == 0) {
        atomicAdd(&sum_y[batch], ssum);
        atomicAdd(&cnt[batch], scnt);
    }
}

// ---------------------------------------------------------------- finalize
__global__ void __launch_bounds__(P) chamfer_final(
        const unsigned int* __restrict__ binmin,
        const float* __restrict__ sum_y,
        const unsigned int* __restrict__ cnt,
        float* __restrict__ out, int n)
{
    __shared__ float red[P / 32];
    const int tid   = threadIdx.x;
    const int batch = blockIdx.x;

    float v = __uint_as_float(binmin[batch * P + tid]);
#pragma unroll
    for (int off = 16; off > 0; off >>= 1) v += __shfl_xor(v, off, 32);
    if ((tid & 31) == 0) red[tid >> 5] = v;
    __syncthreads();

    if (tid == 0) {
        float s = 0.0f;
#pragma unroll
        for (int w = 0; w < P / 32; ++w) s += red[w];
        const float cham_x = s / (float)P;
        const unsigned int c = cnt[batch];
        const float cham_y = sum_y[batch] / (float)(c ? c : 1u);
        atomicAdd(out, (cham_x + cham_y) / (float)n);
    }
}

// ---------------------------------------------------------------- launch
extern "C" void kernel_launch(void* const* d_in, const int* in_sizes, int n_in,
                              void* d_out, int out_size, void* d_ws, size_t ws_size,
                              hipStream_t stream) {
    const float* bins = (const float*)d_in[0];   // [n, 256]
    const float* tp   = (const float*)d_in[1];   // [n, H*W]
    const int n = in_sizes[0] / P;
    const int m = in_sizes[1] / n;

    unsigned int* binmin = (unsigned int*)d_ws;
    float*        sum_y  = (float*)((char*)d_ws + (size_t)n * P * sizeof(unsigned int));
    unsigned int* cnt    = (unsigned int*)((char*)sum_y + (size_t)n * sizeof(float));
    float*        out    = (float*)d_out;

    chamfer_init<<<1, 256, 0, stream>>>(binmin, sum_y, cnt, out, n);

    dim3 grid((m + PIX_PER_BLOCK - 1) / PIX_PER_BLOCK, n);
    chamfer_main<<<grid, P, 0, stream>>>(bins, tp, m, binmin, sum_y, cnt);

    chamfer_final<<<n, P, 0, stream>>>(binmin, sum_y, cnt, out, n);
}